// FeatTimeAttention_17600775979033
// MI455X (gfx1250) — compile-verified
//
#include <hip/hip_runtime.h>

// Problem constants from the reference: N=256, T=128, D_F=32, L=64
#define NN 256
#define TT 128
#define DF 32
#define LF 64
#define KB_STRIDE 66   // pad 64 -> 66 floats: conflict-free banks, 8B-aligned rows
#define G_STRIDE  33   // pad 32 -> 33 floats: conflict-free row reads

typedef __attribute__((ext_vector_type(2))) float v2f;
typedef __attribute__((ext_vector_type(8))) float v8f;

__device__ __forceinline__ float relu_f(float v) { return fmaxf(v, 0.0f); }

// Broadcast from a compile-time-constant lane: v_readlane_b32 -> SGPR (no LDS,
// no dscnt wait), instead of the ds_bpermute that __shfl lowers to.
__device__ __forceinline__ float bcast_lane(float v, int srclane) {
    return __int_as_float(__builtin_amdgcn_readlane(__float_as_int(v), srclane));
}

// v_rcp_f32 + one Newton step: ~1ulp at a fraction of full-div cost.
__device__ __forceinline__ float fast_rcp(float v) {
    float r = __builtin_amdgcn_rcpf(v);
    return fmaf(fmaf(-v, r, 1.0f), r, r);
}

__global__ __launch_bounds__(128) void feat_time_attention_kernel(
    const float* __restrict__ x,        // (N,T,DF)
    const float* __restrict__ latent,   // (N,LF)
    const float* __restrict__ kernel_w, // (DF,LF)
    const float* __restrict__ bias,     // (DF,LF)
    const float* __restrict__ beta,     // (T,)
    float* __restrict__ out)            // (N,LF)
{
    __shared__ float kb_k[DF * KB_STRIDE];      // ~8.4 KB
    __shared__ float kb_b[DF * KB_STRIDE];      // ~8.4 KB
    __shared__ float z_sh[LF];
    __shared__ float beta_sh[TT];
    __shared__ float G_sh[4][DF * G_STRIDE];    // ~16.9 KB (per-wave Gram staging)
    __shared__ float red[4 * LF];               // cross-wave reduction

    const int tid  = threadIdx.x;
    const int wave = tid >> 5;
    const int lane = tid & 31;
    const int n    = blockIdx.x;

    // ---- stage kernel/bias (padded), latent row, beta ----
    #pragma unroll
    for (int i = 0; i < 16; ++i) {
        int idx = i * 128 + tid;                // 2048 elements, coalesced
        int d = idx >> 6, l = idx & 63;
        kb_k[d * KB_STRIDE + l] = kernel_w[idx];
        kb_b[d * KB_STRIDE + l] = bias[idx];
    }
    if (tid < LF) z_sh[tid] = latent[(size_t)n * LF + tid];
    beta_sh[tid] = beta[tid];
    __syncthreads();

    // normalized temporal weights denominator (redundant per thread, trivial)
    float wsum = 0.0f;
    #pragma unroll 8
    for (int t = 0; t < TT; ++t) wsum += fabsf(beta_sh[t]);
    const float winv = 1.0f / wsum;

    float acc0 = 0.0f, acc1 = 0.0f;             // out cols lane, lane+32
    float* Gw = G_sh[wave];

    for (int t = wave; t < TT; t += 4) {
        const float xv = x[((size_t)n * TT + t) * DF + lane];
        if (t + 4 < TT)
            __builtin_prefetch(&x[((size_t)n * TT + (t + 4)) * DF + lane], 0, 0);

        // ---- rhs: r_d = F[d,:] . z   (lane d owns row d) ----
        float r = 0.0f;
        #pragma unroll
        for (int l = 0; l < LF; ++l) {
            float f = relu_f(fmaf(xv, kb_k[lane * KB_STRIDE + l],
                                      kb_b[lane * KB_STRIDE + l]));
            r = fmaf(f, z_sh[l], r);
        }

        // ---- Gram G = F F^T via V_WMMA_F32_16X16X4_F32 (symmetric: skip c10) ----
        // f32 A/B fragment layout: lane%16 = M (resp N), K = (lane/16)*2 + vgpr
        const int i16 = lane & 15;
        const int h2  = (lane >> 4) << 1;       // 0 or 2
        const float xm0 = __shfl(xv, i16, 32);        // row block 0 x value (bpermute)
        const float xm1 = __shfl(xv, 16 + i16, 32);   // row block 1 x value (bpermute)
        const int rowA = i16 * KB_STRIDE;
        const int rowB = (16 + i16) * KB_STRIDE;

        v8f c00 = {}, c01 = {}, c11 = {};
        #pragma unroll
        for (int kk = 0; kk < 16; ++kk) {       // K = 64 in chunks of 4
            const int c = kk * 4 + h2;
            v2f f0, f1;                          // identical A- and B-fragments
            f0.x = relu_f(fmaf(xm0, kb_k[rowA + c],     kb_b[rowA + c]));
            f0.y = relu_f(fmaf(xm0, kb_k[rowA + c + 1], kb_b[rowA + c + 1]));
            f1.x = relu_f(fmaf(xm1, kb_k[rowB + c],     kb_b[rowB + c]));
            f1.y = relu_f(fmaf(xm1, kb_k[rowB + c + 1], kb_b[rowB + c + 1]));
            c00 = __builtin_amdgcn_wmma_f32_16x16x4_f32(false, f0, false, f0, (short)0, c00, false, false);
            c01 = __builtin_amdgcn_wmma_f32_16x16x4_f32(false, f0, false, f1, (short)0, c01, false, false);
            c11 = __builtin_amdgcn_wmma_f32_16x16x4_f32(false, f1, false, f1, (short)0, c11, false, false);
        }

        // ---- scatter C tiles (VGPR j -> row (lane/16)*8+j, col lane%16) ----
        // c01 holds G[rbase+j][16+i16]; write it straight AND transposed to
        // synthesize the skipped c10 tile (G symmetric).
        const int rbase = (lane >> 4) * 8;
        #pragma unroll
        for (int j = 0; j < 8; ++j) {
            Gw[(rbase + j)      * G_STRIDE + i16]          = c00[j];
            Gw[(rbase + j)      * G_STRIDE + 16 + i16]     = c01[j];
            Gw[(16 + i16)       * G_STRIDE + rbase + j]    = c01[j];   // transpose
            Gw[(16 + rbase + j) * G_STRIDE + 16 + i16]     = c11[j];
        }
        asm volatile("s_wait_dscnt 0x0" ::: "memory");  // wave-local LDS fence

        // ---- lane i pulls row i of [G | r] into registers ----
        float a[DF + 1];
        #pragma unroll
        for (int j = 0; j < DF; ++j) a[j] = Gw[lane * G_STRIDE + j];
        a[DF] = r;

        // ---- register Gauss-Jordan (SPD, no pivoting), v_readlane broadcasts ----
        // Pivot-row scaling folded into the elimination factor:
        //   lane k:  a[j] <- pj - (1-pinv)*pj   = pinv*pj      (scaled pivot row)
        //   lane i:  a[j] <- a[j] - (a[k]*pinv)*pj             (eliminated)
        // -> one cndmask per k, single fma per (k,j).
        #pragma unroll
        for (int k = 0; k < DF; ++k) {
            float piv    = bcast_lane(a[k], k);
            float pinv   = fast_rcp(piv);
            float factor = (lane == k) ? (1.0f - pinv) : a[k] * pinv;
            a[k] = fmaf(-factor, piv, a[k]);    // reuse piv; no extra readlane
            #pragma unroll
            for (int j = k + 1; j <= DF; ++j) {
                float pj = bcast_lane(a[j], k);
                a[j] = fmaf(-factor, pj, a[j]);
            }
        }
        const float s = a[DF];                  // lane's solution component

        // ---- o[l] = sum_d s_d * F[d][l]; lane covers l=lane and l=lane+32 ----
        float o0 = 0.0f, o1 = 0.0f;
        #pragma unroll
        for (int d = 0; d < DF; ++d) {
            float sd = bcast_lane(s,  d);
            float xd = bcast_lane(xv, d);
            float f0 = relu_f(fmaf(xd, kb_k[d * KB_STRIDE + lane],
                                       kb_b[d * KB_STRIDE + lane]));
            float f1 = relu_f(fmaf(xd, kb_k[d * KB_STRIDE + 32 + lane],
                                       kb_b[d * KB_STRIDE + 32 + lane]));
            o0 = fmaf(sd, f0, o0);
            o1 = fmaf(sd, f1, o1);
        }

        const float wt = fabsf(beta_sh[t]) * winv;
        acc0 = fmaf(wt, o0, acc0);
        acc1 = fmaf(wt, o1, acc1);
    }

    // ---- cross-wave reduction, single coalesced store of out[n,:] ----
    red[wave * LF + lane]      = acc0;
    red[wave * LF + 32 + lane] = acc1;
    __syncthreads();
    if (tid < LF) {
        float v = red[tid] + red[LF + tid] + red[2 * LF + tid] + red[3 * LF + tid];
        out[(size_t)n * LF + tid] = v;
    }
}

extern "C" void kernel_launch(void* const* d_in, const int* in_sizes, int n_in,
                              void* d_out, int out_size, void* d_ws, size_t ws_size,
                              hipStream_t stream) {
    (void)in_sizes; (void)n_in; (void)out_size; (void)d_ws; (void)ws_size;
    const float* x        = (const float*)d_in[0];
    const float* latent   = (const float*)d_in[1];
    const float* kernel_w = (const float*)d_in[2];
    const float* bias     = (const float*)d_in[3];
    const float* beta     = (const float*)d_in[4];
    float* out            = (float*)d_out;

    feat_time_attention_kernel<<<NN, 128, 0, stream>>>(x, latent, kernel_w, bias, beta, out);
}